// IDEC_Contrastive_Loss_33414845562970
// MI455X (gfx1250) — compile-verified
//
#include <hip/hip_runtime.h>
#include <hip/hip_bf16.h>

typedef __attribute__((ext_vector_type(2))) float v2f;
typedef __attribute__((ext_vector_type(8))) float v8f;

#define BB    8
#define CCH   256
#define HH    128
#define WW    128
#define NCLS  21
#define NOLD  15
#define PLANE (HH*WW)      // 16384
#define LST   257          // LDS class-row stride in k_accum (257 % 64 != 0 -> no systematic bank conflict)
#define MST   258          // LDS mean-row stride in k_final (even, for paired loads; 258 % 64 == 2)

#define THRESH 0.5f
#define MARGIN 1.0f
#define EPSV   1e-6f

// ---------------------------------------------------------------------------
// Workspace layout (floats): [ sum_new 21*256 | sum_old 21*256 | counts 32(int) ]
// ---------------------------------------------------------------------------
__global__ void k_zero(float* ws) {
  const int n = 2 * NCLS * CCH + 32;
  for (int i = blockIdx.x * blockDim.x + threadIdx.x; i < n;
       i += gridDim.x * blockDim.x)
    ws[i] = 0.0f;  // zero bit-pattern also zeroes the int counts
}

// ---------------------------------------------------------------------------
// k_accum: 256 workgroups (one per WGP-ish), 256 threads.
// WG (b, h0..h0+3): computes pseudo-labels for its 512 pixels, then streams
// both feature tensors (coalesced along W) into per-class LDS accumulators
// via ds_add_f32, then flushes once to global fp32 atomics.
// ---------------------------------------------------------------------------
__global__ __launch_bounds__(256)
void k_accum(const int*   __restrict__ labels,
             const float* __restrict__ f_old,
             const float* __restrict__ f_new,
             const float* __restrict__ out_old,
             float* __restrict__ sum_new,
             float* __restrict__ sum_old,
             int*   __restrict__ counts)
{
  __shared__ int   cls[512];
  __shared__ int   lcnt[NCLS];
  __shared__ float lsN[NCLS * LST];
  __shared__ float lsO[NCLS * LST];

  const int t  = threadIdx.x;
  const int wg = blockIdx.x;           // 0..255
  const int b  = wg >> 5;              // 0..7
  const int h0 = (wg & 31) << 2;       // 0,4,...,124

  for (int i = t; i < NCLS * LST; i += 256) { lsN[i] = 0.f; lsO[i] = 0.f; }
  if (t < NCLS) lcnt[t] = 0;
  __syncthreads();

  // ---- pseudo-label: pl = (label==0) ? argmax_k thresh(out_old[k]) : label
  // only the ::4,::4 strided positions of the 512x512 maps are touched.
  for (int p = t; p < 512; p += 256) {
    const int r = p >> 7, w = p & 127;
    const int h = h0 + r;
    const int lab = labels[((b * 512) + (h << 2)) * 512 + (w << 2)];
    int c = lab;
    if (lab == 0) {
      const float* base =
          out_old + ((size_t)(b * NOLD) * 512 + (size_t)(h << 2)) * 512 + (w << 2);
      float bv = base[0]; if (bv < THRESH) bv = 0.f;
      int bi = 0;
      for (int k = 1; k < NOLD; ++k) {
        float v = base[(size_t)k * 512 * 512];
        if (v < THRESH) v = 0.f;
        if (v > bv) { bv = v; bi = k; }   // strict > == argmax first-occurrence
      }
      c = bi;
    }
    cls[p] = c;
    atomicAdd(&lcnt[c], 1);
  }
  __syncthreads();

  // ---- stream features: lane = w (coalesced 128B/wave), scatter by class
  const int w  = t & 127;
  const int ci = t >> 7;               // channel parity 0/1
  const size_t baseOff = (size_t)b * CCH * PLANE;
  for (int r = 0; r < 4; ++r) {
    const int   cl = cls[(r << 7) + w];
    float* dN = &lsN[cl * LST];
    float* dO = &lsO[cl * LST];
    const size_t pix = (size_t)(h0 + r) * WW + w;
    const float* pN = f_new + baseOff + pix;
    const float* pO = f_old + baseOff + pix;
    for (int c = ci; c < CCH; c += 2) {
      if (c + 8 < CCH) {                      // global_prefetch_b8 ahead
        __builtin_prefetch(&pN[(size_t)(c + 8) * PLANE], 0, 3);
        __builtin_prefetch(&pO[(size_t)(c + 8) * PLANE], 0, 3);
      }
      const float vN = pN[(size_t)c * PLANE];
      const float vO = pO[(size_t)c * PLANE];
      unsafeAtomicAdd(&dN[c], vN);            // ds_add_f32
      unsafeAtomicAdd(&dO[c], vO);
    }
  }
  __syncthreads();

  // ---- one global fp32 atomic per (class,channel) per WG
  for (int i = t; i < NCLS * CCH; i += 256) {
    const int k = i >> 8, c = i & 255;
    unsafeAtomicAdd(&sum_new[i], lsN[k * LST + c]);   // global_atomic_add_f32
    unsafeAtomicAdd(&sum_old[i], lsO[k * LST + c]);
  }
  if (t < NCLS) atomicAdd(&counts[t], lcnt[t]);
}

// ---------------------------------------------------------------------------
// k_final: single wave32. means -> d_ap, then Gram G = M M^T with
// V_WMMA_F32_16X16X4_F32 (fp32, 2x2 class tiles x 64 K-steps), then
// d_an^2[i][j] = Q_i + Q_j - 2 G_ij + 2e(R_i - R_j) + C e^2, triplet sum.
// ---------------------------------------------------------------------------
__global__ __launch_bounds__(32)
void k_final(const float* __restrict__ sum_new,
             const float* __restrict__ sum_old,
             const int*   __restrict__ counts,
             float* __restrict__ out)
{
  __shared__ float Ma[32 * MST];      // mean_a, rows 21..31 zero-padded
  __shared__ float G[32 * 33];        // Gram
  __shared__ float Rsh[32], dapsh[32], part[32];
  __shared__ int   pres[32];

  const int l = threadIdx.x;          // lane == class row (l < 21 active)

  const int   cnt   = (l < NCLS) ? counts[l] : 0;
  const float denom = fmaxf((float)cnt, 1.0f);
  float R = 0.f, dap2 = 0.f;
  for (int c = 0; c < CCH; ++c) {
    float ma = 0.f, mp = 0.f;
    if (l < NCLS) {
      ma = sum_new[l * CCH + c] / denom;
      mp = sum_old[l * CCH + c] / denom;
    }
    Ma[l * MST + c] = ma;
    R += ma;
    const float d = ma - mp + EPSV;
    dap2 += d * d;
  }
  Rsh[l]   = R;
  dapsh[l] = sqrtf(dap2);
  pres[l]  = (l >= 1 && l < NCLS && cnt > 0) ? 1 : 0;
  __syncthreads();

  // A 16x4 f32 layout: lane L -> row m=L%16, VGPR{0,1} -> K = (L/16)*2 + {0,1}.
  // B 4x16 layout: lane L -> col n=L%16, VGPR{0,1} -> K = (L/16)*2 + {0,1}.
  // => for G = M M^T the per-lane B vector of class-block j equals the
  //    per-lane A vector of class-block j. Full EXEC, no divergence here.
  const int m     = l & 15;
  const int khalf = l >> 4;
  v8f g00 = {0.f,0.f,0.f,0.f,0.f,0.f,0.f,0.f};
  v8f g01 = g00, g10 = g00, g11 = g00;
  for (int kb = 0; kb < 64; ++kb) {
    const int k0 = (kb << 2) + (khalf << 1);
    v2f a0, a1;
    a0.x = Ma[m * MST + k0];          a0.y = Ma[m * MST + k0 + 1];
    a1.x = Ma[(16 + m) * MST + k0];   a1.y = Ma[(16 + m) * MST + k0 + 1];
    g00 = __builtin_amdgcn_wmma_f32_16x16x4_f32(false, a0, false, a0, (short)0, g00, false, false);
    g01 = __builtin_amdgcn_wmma_f32_16x16x4_f32(false, a0, false, a1, (short)0, g01, false, false);
    g10 = __builtin_amdgcn_wmma_f32_16x16x4_f32(false, a1, false, a0, (short)0, g10, false, false);
    g11 = __builtin_amdgcn_wmma_f32_16x16x4_f32(false, a1, false, a1, (short)0, g11, false, false);
  }
  // C/D layout: lane L, VGPR v -> D[v + (L/16)*8][L%16]
  for (int v = 0; v < 8; ++v) {
    const int gi = v + (khalf << 3);
    G[gi * 33 + m]             = g00[v];
    G[gi * 33 + 16 + m]        = g01[v];
    G[(16 + gi) * 33 + m]      = g10[v];
    G[(16 + gi) * 33 + 16 + m] = g11[v];
  }
  __syncthreads();

  float acc = 0.f;
  if (l < NCLS && pres[l]) {
    const float Qi = G[l * 33 + l];
    const float Ri = Rsh[l];
    const float di = dapsh[l];
    for (int j = 0; j < NCLS; ++j) {
      if (j == l || !pres[j]) continue;
      float dan2 = Qi + G[j * 33 + j] - 2.f * G[l * 33 + j]
                 + 2.f * EPSV * (Ri - Rsh[j]) + (float)CCH * (EPSV * EPSV);
      dan2 = fmaxf(dan2, 0.f);
      const float tr = di - sqrtf(dan2) + MARGIN;
      if (tr > 0.f) acc += tr;
    }
  }
  part[l] = acc;
  __syncthreads();

  if (l == 0) {
    float tot = 0.f; int Kp = 0;
    for (int i = 0; i < 32; ++i) { tot += part[i]; Kp += pres[i]; }
    const float kf = (float)Kp;
    out[0] = (Kp > 1) ? tot / fmaxf(kf * (kf - 1.f), 1.f) : 0.f;
  }
}

// ---------------------------------------------------------------------------
extern "C" void kernel_launch(void* const* d_in, const int* in_sizes, int n_in,
                              void* d_out, int out_size, void* d_ws, size_t ws_size,
                              hipStream_t stream) {
  const int*   labels  = (const int*)d_in[0];     // (8,512,512) int
  const float* f_old   = (const float*)d_in[1];   // (8,256,128,128)
  const float* f_new   = (const float*)d_in[2];   // (8,256,128,128)
  const float* out_old = (const float*)d_in[3];   // (8,15,512,512)
  // d_in[4] outputs, d_in[5] prototypes: unused by the reference.

  float* ws      = (float*)d_ws;
  float* sum_new = ws;
  float* sum_old = ws + NCLS * CCH;
  int*   counts  = (int*)(ws + 2 * NCLS * CCH);

  k_zero <<<16, 256, 0, stream>>>(ws);
  k_accum<<<256, 256, 0, stream>>>(labels, f_old, f_new, out_old,
                                   sum_new, sum_old, counts);
  k_final<<<1, 32, 0, stream>>>(sum_new, sum_old, counts, (float*)d_out);
}